// TurboFlashAttention_2757369004770
// MI455X (gfx1250) — compile-verified
//
#include <hip/hip_runtime.h>

typedef float v2f __attribute__((ext_vector_type(2)));
typedef float v8f __attribute__((ext_vector_type(8)));

#define KSTRIDE 68   // K tile LDS row stride (floats): banks 4*ln+d -> conflict-free b64 reads
#define VSTRIDE 72   // V tile LDS row stride (floats): half-waves hit disjoint bank ranges

__device__ __forceinline__ float bperm(float v, int srcLane) {
    // wave32 lane gather; index in bytes
    return __int_as_float(__builtin_amdgcn_ds_bpermute(srcLane << 2, __float_as_int(v)));
}

static __device__ __forceinline__ v8f wmma_f32(v2f a, v2f b, v8f c) {
    // D(16x16,f32) = A(16x4,f32) * B(4x16,f32) + C
    return __builtin_amdgcn_wmma_f32_16x16x4_f32(false, a, false, b, (short)0, c, false, false);
}

__global__ void __launch_bounds__(256) zero_fill4(float4* __restrict__ p, long n4) {
    long i = (long)blockIdx.x * blockDim.x + threadIdx.x;
    long stride = (long)gridDim.x * blockDim.x;
    float4 z = make_float4(0.f, 0.f, 0.f, 0.f);
    for (; i < n4; i += stride) p[i] = z;
}

// One 16-key tile step of online-softmax flash attention for one wave.
// S^T = K*Q^T (columns = queries -> per-lane softmax stats), O^T += V^T * P^T.
template <bool MASKED>
static __device__ __forceinline__ void fa_tile(const float* __restrict__ shK,
                                               const float* __restrict__ shV,
                                               const v2f*   __restrict__ bq,
                                               v8f*         __restrict__ acc,
                                               float& m, float& l,
                                               int kbase, int q_abs, int hi, int ln, int lane)
{
    const float NEG_INF = -__builtin_inff();

    // ---- S^T = K * Q^T : two parallel 8-deep k=4 fp32 WMMA chains over D=64 ----
    v8f st0 = (v8f){0.f,0.f,0.f,0.f,0.f,0.f,0.f,0.f};
    v8f st1 = (v8f){0.f,0.f,0.f,0.f,0.f,0.f,0.f,0.f};
#pragma unroll
    for (int c = 0; c < 8; ++c) {
        int d0 = 4 * c + 2 * hi;
        int d1 = 4 * (c + 8) + 2 * hi;
        v2f ak0 = *(const v2f*)(shK + ln * KSTRIDE + d0);
        v2f ak1 = *(const v2f*)(shK + ln * KSTRIDE + d1);
        st0 = wmma_f32(ak0, bq[c],     st0);
        st1 = wmma_f32(ak1, bq[c + 8], st1);
    }
    v8f st = st0 + st1;

    // ---- (masked) online softmax; per-lane column = query ----
    float sv[8];
    float mt = NEG_INF;
#pragma unroll
    for (int r = 0; r < 8; ++r) {
        float x = st[r];
        if (MASKED) {
            int key = kbase + r + 8 * hi;      // row index of S^T
            x = (key <= q_abs) ? x : NEG_INF;
        }
        sv[r] = x;
        mt = fmaxf(mt, x);
    }
    mt = fmaxf(mt, bperm(mt, lane ^ 16));      // merge key halves
    float mnew  = fmaxf(m, mt);
    float alpha = __expf(m - mnew);            // exp(-inf)=0 on first tile

    v8f pt;
    float rs = 0.f;
#pragma unroll
    for (int r = 0; r < 8; ++r) {
        float pv = __expf(sv[r] - mnew);       // masked -> 0
        pt[r] = pv;
        rs += pv;
    }
    rs += bperm(rs, lane ^ 16);
    l = l * alpha + rs;
    m = mnew;
#pragma unroll
    for (int dc = 0; dc < 4; ++dc) acc[dc] *= alpha;

    // ---- re-lay P^T (C layout) into B-layout frags: bp[c] covers keys 4c..4c+3 ----
    // source of P^T[kbase+kk][q]: VGPR pt[kk&7] at lane = q + 16*(kk>>3)
    v2f bp[4];
#pragma unroll
    for (int c = 0; c < 4; ++c) {
        const int k0lo = 4 * c + 0, k1lo = 4 * c + 1;   // lanes with hi==0
        const int k0hi = 4 * c + 2, k1hi = 4 * c + 3;   // lanes with hi==1
        float x_lo = bperm(pt[k0lo & 7], ln + ((k0lo >> 3) << 4));
        float x_hi = bperm(pt[k0hi & 7], ln + ((k0hi >> 3) << 4));
        float y_lo = bperm(pt[k1lo & 7], ln + ((k1lo >> 3) << 4));
        float y_hi = bperm(pt[k1hi & 7], ln + ((k1hi >> 3) << 4));
        v2f t;
        t.x = hi ? x_hi : x_lo;
        t.y = hi ? y_hi : y_lo;
        bp[c] = t;
    }

    // ---- O^T += V^T * P^T : A frag = V^T (M = d, K = key chunk); 4 parallel chains ----
#pragma unroll
    for (int dc = 0; dc < 4; ++dc) {
        const int dd = 16 * dc + ln;
#pragma unroll
        for (int c = 0; c < 4; ++c) {
            const int kk = 4 * c + 2 * hi;
            v2f av;
            av.x = shV[kk       * VSTRIDE + dd];
            av.y = shV[(kk + 1) * VSTRIDE + dd];
            acc[dc] = wmma_f32(av, bp[c], acc[dc]);
        }
    }
}

// 4 waves / workgroup; each wave owns one 16-query tile; waves share LDS-staged K/V tiles.
__global__ void __launch_bounds__(128) fa_fwd(const float* __restrict__ Q,
                                              const float* __restrict__ K,
                                              const float* __restrict__ V,
                                              float* __restrict__ O)
{
    constexpr int S = 2048, D = 64;
    const int wave = threadIdx.x >> 5;        // 0..3
    const int lane = threadIdx.x & 31;
    const int hi   = lane >> 4;
    const int ln   = lane & 15;
    const int qt   = blockIdx.x * 4 + wave;   // this wave's query tile
    const int qtMax = blockIdx.x * 4 + 3;
    const int bh   = blockIdx.y;

    const float scale = 0.125f;               // 1/sqrt(64)

    const float* __restrict__ qb = Q + ((long)bh * S + (long)qt * 16) * D;
    const float* __restrict__ kb = K + (long)bh * S * D;
    const float* __restrict__ vb = V + (long)bh * S * D;

    __shared__ float shK[16 * KSTRIDE];
    __shared__ float shV[16 * VSTRIDE];

    // Q^T fragments (B operand of S^T = K*Q^T), scale folded in.
    v2f bq[16];
#pragma unroll
    for (int c = 0; c < 16; ++c) {
        int d = 4 * c + 2 * hi;
        v2f t;
        t.x = qb[ln * D + d]     * scale;
        t.y = qb[ln * D + d + 1] * scale;
        bq[c] = t;
    }

    v8f acc[4];
#pragma unroll
    for (int dc = 0; dc < 4; ++dc) acc[dc] = (v8f){0.f,0.f,0.f,0.f,0.f,0.f,0.f,0.f};

    float m = -__builtin_inff();
    float l = 0.f;
    const int q_abs = qt * 16 + ln;

    const int t = threadIdx.x;                // staging: 2 float4 per tensor per thread
    const int row0 = t >> 4,          c40 = t & 15;
    const int row1 = (t + 128) >> 4,  c41 = (t + 128) & 15;

    for (int kt = 0; kt <= qtMax; ++kt) {
        const int kbase = kt * 16;

        // coalesced global -> regs (issue before barrier to overlap latency)
        float4 k0 = *(const float4*)(kb + (long)(kbase + row0) * D + c40 * 4);
        float4 k1 = *(const float4*)(kb + (long)(kbase + row1) * D + c41 * 4);
        float4 v0 = *(const float4*)(vb + (long)(kbase + row0) * D + c40 * 4);
        float4 v1 = *(const float4*)(vb + (long)(kbase + row1) * D + c41 * 4);

        __syncthreads();   // previous tile's compute done before overwrite
        *(float4*)(shK + row0 * KSTRIDE + c40 * 4) = k0;
        *(float4*)(shK + row1 * KSTRIDE + c41 * 4) = k1;
        *(float4*)(shV + row0 * VSTRIDE + c40 * 4) = v0;
        *(float4*)(shV + row1 * VSTRIDE + c41 * 4) = v1;
        __syncthreads();   // staged tile visible to all waves

        if (kt < qt) {
            fa_tile<false>(shK, shV, bq, acc, m, l, kbase, q_abs, hi, ln, lane);
        } else if (kt == qt) {
            fa_tile<true>(shK, shV, bq, acc, m, l, kbase, q_abs, hi, ln, lane);
        }
        // waves with qt < kt only participate in staging/barriers
    }

    // ---- epilogue: O[q][d] = acc^T / l ----
    const float inv_l = 1.f / l;
    float* __restrict__ ob = O + ((long)bh * S + (long)qt * 16 + ln) * D;
#pragma unroll
    for (int dc = 0; dc < 4; ++dc) {
#pragma unroll
        for (int r = 0; r < 8; ++r) {
            ob[16 * dc + r + 8 * hi] = acc[dc][r] * inv_l;
        }
    }
}

extern "C" void kernel_launch(void* const* d_in, const int* in_sizes, int n_in,
                              void* d_out, int out_size, void* d_ws, size_t ws_size,
                              hipStream_t stream) {
    (void)in_sizes; (void)n_in; (void)out_size; (void)d_ws; (void)ws_size;

    const float* q = (const float*)d_in[0];
    const float* k = (const float*)d_in[1];
    const float* v = (const float*)d_in[2];
    float* out = (float*)d_out;

    constexpr long B = 2, H = 16, S = 2048, D = 64;
    constexpr long o_elems = B * H * S * D;       // 4,194,304
    constexpr long w_elems = B * H * S * S;       // 134,217,728

    // Zero the attn_weights placeholder region (reference returns zeros).
    float* attnw = out + o_elems;
    zero_fill4<<<4096, 256, 0, stream>>>((float4*)attnw, w_elems / 4);

    // Flash attention: 4-wave workgroups sharing LDS-staged K/V tiles.
    dim3 grid((unsigned)(S / 64), (unsigned)(B * H));
    fa_fwd<<<grid, 128, 0, stream>>>(q, k, v, out);
}